// VQVAE_47218870452303
// MI455X (gfx1250) — compile-verified
//
#include <hip/hip_runtime.h>
#include <hip/hip_bf16.h>

// ---------------- problem constants ----------------
#define NROWS   524288
#define D_IN    14
#define HID     128
#define LAT     5
#define VOCAB   256
#define BETA_F  0.25f

#define ROWS_PB 64            // rows per block
#define TPB     128           // 4 wave32s
#define HSTR    129           // padded LDS row stride (bank-conflict-free column scans)

// flat output offsets (f32 elements)
#define OFF_OUT   0LL
#define OFF_LOSS  ((long long)NROWS * D_IN)
#define OFF_PERP  (OFF_LOSS + 1)
#define OFF_OH    (OFF_LOSS + 2)
#define OFF_IDX   (OFF_OH + (long long)NROWS * VOCAB)

// workspace layout (floats) when large enough: [0,16384) swizzled W2, then hist[256]+loss
#define W2S_FLOATS (HID * HID)

typedef float v2f __attribute__((ext_vector_type(2)));
typedef float v8f __attribute__((ext_vector_type(8)));

static __device__ __forceinline__ v8f wmma_f32(v2f a, v2f b, v8f c) {
  // V_WMMA_F32_16X16X4_F32 : D = A(16x4, f32) * B(4x16, f32) + C(16x16, f32)
  return __builtin_amdgcn_wmma_f32_16x16x4_f32(
      /*neg_a=*/false, a, /*neg_b=*/false, b,
      /*c_mod=*/(short)0, c, /*reuse_a=*/false, /*reuse_b=*/false);
}

// ---------------- ws init ----------------
__global__ void vq_init(float* __restrict__ hist) {
  int t = threadIdx.x;
  if (t < VOCAB + 1) hist[t] = 0.0f;   // [0..255] histogram, [256] loss accumulator
}

// ---------------- one-time W2 pair-swizzle into workspace ----------------
// w2s[(k/2)*256 + 2n + (k&1)] = W2[k][n]  -> B-fragment = one aligned b64 load
__global__ void w2_swizzle(const float* __restrict__ ew2, float* __restrict__ w2s) {
  const int i = blockIdx.x * 256 + threadIdx.x;   // 16384 elements
  const int k = i >> 7, n = i & 127;
  w2s[(k >> 1) * 256 + n * 2 + (k & 1)] = ew2[i];
}

// ---------------- fused main kernel ----------------
template <bool SWZ>
__global__ __launch_bounds__(TPB) void vqvae_main(
    const float* __restrict__ x,
    const float* __restrict__ ew1, const float* __restrict__ eb1,
    const float* __restrict__ w2,  const float* __restrict__ eb2,   // swizzled if SWZ
    const float* __restrict__ ew3, const float* __restrict__ eb3,
    const float* __restrict__ emb,
    const float* __restrict__ dw1, const float* __restrict__ db1,
    const float* __restrict__ dw2, const float* __restrict__ db2,
    float* __restrict__ out, float* __restrict__ hist) {

  __shared__ float sX[ROWS_PB * 16];        //  4 KB  (D_IN padded to 16)
  __shared__ float sW1[8 * 256];            //  8 KB  pair-swizzled (K padded 14->16)
  __shared__ float sH[ROWS_PB * HSTR];      // ~32 KB activations (wave-private strips)
  __shared__ float sQ[ROWS_PB * 8];         //  2 KB  quantized latents (LAT padded to 8)
  __shared__ float sDW1[4 * 256];           //  4 KB  pair-swizzled (K padded 5->8)
  __shared__ float sDW2[64 * 32];           //  8 KB  pair-swizzled (N padded 14->16)
  __shared__ int   sIdx[ROWS_PB];
  __shared__ int   sHist[VOCAB];
  __shared__ float sRed[TPB];

  const int tid  = threadIdx.x;
  const int lane = tid & 31;
  const int wave = tid >> 5;
  const int l16  = lane & 15;
  const int lHi  = (lane >> 4) & 1;        // 0 or 1
  const int m0   = wave * 16;              // wave's M-strip within the block tile
  const long long row0 = (long long)blockIdx.x * ROWS_PB;

  // ---- stage (zero-padded, pair-swizzled) operands into LDS ----
  for (int i = tid; i < ROWS_PB * 16; i += TPB) {
    int r = i >> 4, c = i & 15;
    sX[i] = (c < D_IN) ? __builtin_nontemporal_load(&x[(size_t)(row0 + r) * D_IN + c]) : 0.0f;
  }
  for (int i = tid; i < 16 * HID; i += TPB) {          // enc_w1: 16x128 (rows 14,15 = 0)
    int k = i >> 7, n = i & 127;
    sW1[(k >> 1) * 256 + n * 2 + (k & 1)] = (k < D_IN) ? ew1[k * HID + n] : 0.0f;
  }
  for (int i = tid; i < 8 * HID; i += TPB) {           // dec_w1: 8x128 (rows 5..7 = 0)
    int k = i >> 7, n = i & 127;
    sDW1[(k >> 1) * 256 + n * 2 + (k & 1)] = (k < LAT) ? dw1[k * HID + n] : 0.0f;
  }
  for (int i = tid; i < HID * 16; i += TPB) {          // dec_w2: 128x16 (cols 14,15 = 0)
    int k = i >> 4, n = i & 15;
    sDW2[(k >> 1) * 32 + n * 2 + (k & 1)] = (n < D_IN) ? dw2[k * D_IN + n] : 0.0f;
  }
  for (int b = tid; b < VOCAB; b += TPB) sHist[b] = 0;
  __syncthreads();

  const v8f vzero = {0.f, 0.f, 0.f, 0.f, 0.f, 0.f, 0.f, 0.f};
  v8f acc[8];

  // ================= encoder layer 1: X(64x16) @ W1(16x128) -> relu -> sH =================
#pragma unroll
  for (int nt = 0; nt < 8; ++nt) acc[nt] = vzero;
#pragma unroll
  for (int k0 = 0; k0 < 16; k0 += 4) {
    const int k = k0 + 2 * lHi;
    const v2f a = *(const v2f*)&sX[(m0 + l16) * 16 + k];           // aligned b64
#pragma unroll
    for (int nt = 0; nt < 8; ++nt) {
      const v2f b = *(const v2f*)&sW1[((k0 >> 1) + lHi) * 256 + (nt * 16 + l16) * 2];
      acc[nt] = wmma_f32(a, b, acc[nt]);
    }
  }
#pragma unroll
  for (int nt = 0; nt < 8; ++nt) {
    const float bias = eb1[nt * 16 + l16];
#pragma unroll
    for (int v = 0; v < 8; ++v) {
      float val = acc[nt][v] + bias;
      sH[(m0 + v + 8 * lHi) * HSTR + nt * 16 + l16] = fmaxf(val, 0.0f);
    }
  }

  // ================= encoder layer 2: H(64x128) @ W2(128x128) -> relu -> sH =================
  // B-fragments streamed from global (L2-resident 64 KB); swizzled path = 1 b64 load each.
#pragma unroll
  for (int nt = 0; nt < 8; ++nt) acc[nt] = vzero;
  for (int k0 = 0; k0 < HID; k0 += 4) {
    const int k = k0 + 2 * lHi;
    v2f a;
    a.x = sH[(m0 + l16) * HSTR + k];
    a.y = sH[(m0 + l16) * HSTR + k + 1];
#pragma unroll
    for (int nt = 0; nt < 8; ++nt) {
      const int n = nt * 16 + l16;
      v2f b;
      if constexpr (SWZ) {
        b = *(const v2f*)&w2[((k0 >> 1) + lHi) * 256 + n * 2];
      } else {
        b.x = w2[k * HID + n];
        b.y = w2[(k + 1) * HID + n];
      }
      acc[nt] = wmma_f32(a, b, acc[nt]);
    }
  }
#pragma unroll
  for (int nt = 0; nt < 8; ++nt) {
    const float bias = eb2[nt * 16 + l16];
#pragma unroll
    for (int v = 0; v < 8; ++v) {
      float val = acc[nt][v] + bias;
      sH[(m0 + v + 8 * lHi) * HSTR + nt * 16 + l16] = fmaxf(val, 0.0f);
    }
  }
  __syncthreads();   // VQ phase reads all rows of sH

  // ================= latent projection + VQ (one thread per row) =================
  float rowLoss = 0.0f;
  if (tid < ROWS_PB) {
    const int r = tid;
    float q[LAT];
#pragma unroll
    for (int j = 0; j < LAT; ++j) q[j] = eb3[j];
    for (int k = 0; k < HID; ++k) {           // stride 129 -> conflict-free column scan
      const float h = sH[r * HSTR + k];
#pragma unroll
      for (int j = 0; j < LAT; ++j) q[j] = fmaf(h, ew3[k * LAT + j], q[j]);
    }
    float best = 3.4e38f; int bi = 0;
    for (int c = 0; c < VOCAB; ++c) {         // uniform addresses -> scalar loads
      float d = 0.0f;
#pragma unroll
      for (int j = 0; j < LAT; ++j) {
        const float df = q[j] - emb[c * LAT + j];
        d = fmaf(df, df, d);
      }
      if (d < best) { best = d; bi = c; }
    }
    sIdx[r] = bi;
    atomicAdd(&sHist[bi], 1);
    rowLoss = best;                            // sum_j (q - e_best)^2
#pragma unroll
    for (int j = 0; j < LAT; ++j) sQ[r * 8 + j] = emb[bi * LAT + j];
    sQ[r * 8 + 5] = 0.f; sQ[r * 8 + 6] = 0.f; sQ[r * 8 + 7] = 0.f;
    __builtin_nontemporal_store((float)bi, &out[OFF_IDX + row0 + r]);  // indices
  }
  sRed[tid] = rowLoss;
  __syncthreads();   // sQ/sIdx ready; sH free to overwrite

  // ---- block loss -> global accumulator; histogram -> global ----
  if (tid == 0) {
    float s = 0.0f;
    for (int i = 0; i < ROWS_PB; ++i) s += sRed[i];
    atomicAdd(&hist[VOCAB], s);
  }
  for (int b = tid; b < VOCAB; b += TPB) {
    const int c = sHist[b];
    if (c) atomicAdd(&hist[b], (float)c);
  }

  // ---- one-hot min_encodings: dominant traffic, coalesced non-temporal float2 ----
  {
    float* oneHot = out + OFF_OH;              // base 8-byte aligned
    const int c = tid * 2;
    for (int r = 0; r < ROWS_PB; ++r) {
      const int id = sIdx[r];
      v2f val; val.x = (c == id) ? 1.0f : 0.0f; val.y = (c + 1 == id) ? 1.0f : 0.0f;
      __builtin_nontemporal_store(val, (v2f*)(oneHot + (size_t)(row0 + r) * VOCAB + c));
    }
  }

  // ================= decoder layer 1: Q(64x8) @ dW1(8x128) -> relu -> sH =================
#pragma unroll
  for (int nt = 0; nt < 8; ++nt) acc[nt] = vzero;
#pragma unroll
  for (int k0 = 0; k0 < 8; k0 += 4) {
    const int k = k0 + 2 * lHi;
    const v2f a = *(const v2f*)&sQ[(m0 + l16) * 8 + k];            // aligned b64
#pragma unroll
    for (int nt = 0; nt < 8; ++nt) {
      const v2f b = *(const v2f*)&sDW1[((k0 >> 1) + lHi) * 256 + (nt * 16 + l16) * 2];
      acc[nt] = wmma_f32(a, b, acc[nt]);
    }
  }
#pragma unroll
  for (int nt = 0; nt < 8; ++nt) {
    const float bias = db1[nt * 16 + l16];
#pragma unroll
    for (int v = 0; v < 8; ++v) {
      float val = acc[nt][v] + bias;
      sH[(m0 + v + 8 * lHi) * HSTR + nt * 16 + l16] = fmaxf(val, 0.0f);
    }
  }

  // ================= decoder layer 2: H(64x128) @ dW2(128x16pad) -> output =================
  v8f o = vzero;
  for (int k0 = 0; k0 < HID; k0 += 4) {
    const int k = k0 + 2 * lHi;
    v2f a;
    a.x = sH[(m0 + l16) * HSTR + k];
    a.y = sH[(m0 + l16) * HSTR + k + 1];
    const v2f b = *(const v2f*)&sDW2[((k0 >> 1) + lHi) * 32 + l16 * 2];
    o = wmma_f32(a, b, o);
  }
  if (l16 < D_IN) {
    const float bias = db2[l16];
#pragma unroll
    for (int v = 0; v < 8; ++v) {
      const long long grow = row0 + m0 + v + 8 * lHi;
      __builtin_nontemporal_store(o[v] + bias, &out[OFF_OUT + grow * D_IN + l16]);
    }
  }
}

// ---------------- scalar finalize: loss + perplexity ----------------
__global__ void vq_finalize(float* __restrict__ out, const float* __restrict__ hist) {
  __shared__ float red[VOCAB];
  const int t = threadIdx.x;
  const float em = hist[t] * (1.0f / (float)NROWS);
  red[t] = em * logf(em + 1e-10f);
  __syncthreads();
  for (int s = VOCAB / 2; s > 0; s >>= 1) {
    if (t < s) red[t] += red[t + s];
    __syncthreads();
  }
  if (t == 0) {
    out[OFF_LOSS] = hist[VOCAB] * ((1.0f + BETA_F) / ((float)NROWS * (float)LAT));
    out[OFF_PERP] = expf(-red[0]);
  }
}

extern "C" void kernel_launch(void* const* d_in, const int* in_sizes, int n_in,
                              void* d_out, int out_size, void* d_ws, size_t ws_size,
                              hipStream_t stream) {
  const float* x    = (const float*)d_in[0];
  const float* ew1  = (const float*)d_in[1];
  const float* eb1  = (const float*)d_in[2];
  const float* ew2  = (const float*)d_in[3];
  const float* eb2  = (const float*)d_in[4];
  const float* ew3  = (const float*)d_in[5];
  const float* eb3  = (const float*)d_in[6];
  const float* emb  = (const float*)d_in[7];
  const float* dw1  = (const float*)d_in[8];
  const float* db1  = (const float*)d_in[9];
  const float* dw2  = (const float*)d_in[10];
  const float* db2  = (const float*)d_in[11];
  float* out = (float*)d_out;
  float* wsF = (float*)d_ws;

  const bool swz = ws_size >= (size_t)(W2S_FLOATS + VOCAB + 1) * sizeof(float);
  float* hist = swz ? (wsF + W2S_FLOATS) : wsF;   // hist[256] + loss accumulator

  vq_init<<<1, 288, 0, stream>>>(hist);
  if (swz) {
    w2_swizzle<<<(HID * HID) / 256, 256, 0, stream>>>(ew2, wsF);
    vqvae_main<true><<<NROWS / ROWS_PB, TPB, 0, stream>>>(
        x, ew1, eb1, /*w2=*/wsF, eb2, ew3, eb3, emb, dw1, db1, dw2, db2, out, hist);
  } else {
    vqvae_main<false><<<NROWS / ROWS_PB, TPB, 0, stream>>>(
        x, ew1, eb1, /*w2=*/ew2, eb2, ew3, eb3, emb, dw1, db1, dw2, db2, out, hist);
  }
  vq_finalize<<<1, VOCAB, 0, stream>>>(out, hist);
}